// CrossAttention_86165633892747
// MI455X (gfx1250) — compile-verified
//
#include <hip/hip_runtime.h>
#include <hip/hip_bf16.h>

typedef __attribute__((ext_vector_type(2))) float v2f;
typedef __attribute__((ext_vector_type(8))) float v8f;

#define DIM 16

// ---------------------------------------------------------------------------
// Prep kernel (one tiny block): fold the two linear layers.
//   M[n][k]  = sum_j Wout[n][j] * Wv[j][k]          (16x16)
//   c[n]     = sum_j Wout[n][j] * bv[j] + bout[n]   (16)
// M is stored pre-swizzled in V_WMMA_F32_16X16X4_F32 B-fragment order:
//   chunk c (K base 4c), lane L:  k0 = 4c + (L>=16 ? 2 : 0), n = L&15
//   wsB[(c*32+L)*2 + {0,1}] = M_T[k0][n], M_T[k0+1][n]  (B[k][n] = M[n][k])
// ---------------------------------------------------------------------------
__global__ void attn_prep_kernel(const float* __restrict__ in_w,
                                 const float* __restrict__ in_b,
                                 const float* __restrict__ out_w,
                                 const float* __restrict__ out_b,
                                 float* __restrict__ wsB,
                                 float* __restrict__ wsC) {
    const int t = threadIdx.x;
    const float* Wv = in_w + 2 * DIM * DIM;   // rows [2*DIM, 3*DIM)
    const float* bv = in_b + 2 * DIM;

    if (t < 128) {                             // 4 chunks * 32 lanes
        const int c    = t >> 5;
        const int lane = t & 31;
        const int n    = lane & 15;
        const int k0   = 4 * c + ((lane >> 4) ? 2 : 0);
        float s0 = 0.0f, s1 = 0.0f;
        #pragma unroll
        for (int j = 0; j < DIM; ++j) {
            const float w = out_w[n * DIM + j];
            s0 += w * Wv[j * DIM + k0];
            s1 += w * Wv[j * DIM + k0 + 1];
        }
        wsB[t * 2 + 0] = s0;
        wsB[t * 2 + 1] = s1;
    }
    if (t < DIM) {                             // folded bias
        float s = out_b[t];
        #pragma unroll
        for (int j = 0; j < DIM; ++j) s += out_w[t * DIM + j] * bv[j];
        wsC[t] = s;
    }
}

// ---------------------------------------------------------------------------
// Main kernel: one 16-row tile per wave.
//   D(16x16) = chem_tile(16x16) x B(16x16) + bias, via 4x V_WMMA_F32_16X16X4_F32
// Purely HBM-bound: 8B load x4 + 4B store x8 per lane, 4 WMMA per wave.
// ---------------------------------------------------------------------------
__global__ __launch_bounds__(256) void attn_main_kernel(
        const float* __restrict__ chem,
        const float* __restrict__ wsB,
        const float* __restrict__ wsC,
        float* __restrict__ out,
        long n_tiles) {
    const int  lane   = threadIdx.x & 31;
    const int  waveId = threadIdx.x >> 5;
    const long tile   = (long)blockIdx.x * 8 + waveId;
    if (tile >= n_tiles) return;               // wave-uniform: EXEC stays all-1s

    const int  half    = lane >> 4;            // 0: lanes 0-15, 1: lanes 16-31
    const int  l15     = lane & 15;
    const long rowBase = tile * 16;

    // B fragments (folded weight, pre-swizzled; L2-resident)
    v2f b0 = *(const v2f*)(wsB + (0 * 32 + lane) * 2);
    v2f b1 = *(const v2f*)(wsB + (1 * 32 + lane) * 2);
    v2f b2 = *(const v2f*)(wsB + (2 * 32 + lane) * 2);
    v2f b3 = *(const v2f*)(wsB + (3 * 32 + lane) * 2);

    // C initialized to folded bias: D[m][n] layout -> bias depends only on n = lane&15
    const float cb = wsC[l15];
    v8f acc = {cb, cb, cb, cb, cb, cb, cb, cb};

    // A fragments: lane<16 -> chem[row=l15][4c+0..1]; lane>=16 -> chem[row=l15][4c+2..3]
    const float* abase = chem + (rowBase + l15) * DIM + 2 * half;
    v2f a0 = *(const v2f*)(abase + 0);
    v2f a1 = *(const v2f*)(abase + 4);
    v2f a2 = *(const v2f*)(abase + 8);
    v2f a3 = *(const v2f*)(abase + 12);

    // D = A x B + C over K = 0..15 in chunks of 4 (exact fp32 path)
    acc = __builtin_amdgcn_wmma_f32_16x16x4_f32(false, a0, false, b0, (short)0, acc, false, false);
    acc = __builtin_amdgcn_wmma_f32_16x16x4_f32(false, a1, false, b1, (short)0, acc, false, false);
    acc = __builtin_amdgcn_wmma_f32_16x16x4_f32(false, a2, false, b2, (short)0, acc, false, false);
    acc = __builtin_amdgcn_wmma_f32_16x16x4_f32(false, a3, false, b3, (short)0, acc, false, false);

    // Store: accumulator VGPR g -> row (g + 8*half), col l15
    float* orow = out + (rowBase + 8 * half) * DIM + l15;
    #pragma unroll
    for (int g = 0; g < 8; ++g)
        orow[g * DIM] = acc[g];
}

extern "C" void kernel_launch(void* const* d_in, const int* in_sizes, int n_in,
                              void* d_out, int out_size, void* d_ws, size_t ws_size,
                              hipStream_t stream) {
    // Input order (setup_inputs): fp_16, chem_16, in_proj_weight, in_proj_bias,
    //                             out_proj_weight, out_proj_bias
    // fp_16 is mathematically dead (softmax over a singleton axis == 1).
    const float* chem  = (const float*)d_in[1];
    const float* in_w  = (const float*)d_in[2];
    const float* in_b  = (const float*)d_in[3];
    const float* out_w = (const float*)d_in[4];
    const float* out_b = (const float*)d_in[5];

    float* wsB = (float*)d_ws;        // 256 floats: B fragments
    float* wsC = wsB + 256;           // 16 floats: folded bias

    const long rows    = (long)in_sizes[1] / DIM;   // B = 2097152
    const long n_tiles = rows / 16;                 // 16 rows per wave-tile

    attn_prep_kernel<<<1, 128, 0, stream>>>(in_w, in_b, out_w, out_b, wsB, wsC);

    const int blocks = (int)((n_tiles + 7) / 8);    // 8 waves per 256-thread block
    attn_main_kernel<<<blocks, 256, 0, stream>>>(chem, wsB, wsC, (float*)d_out, n_tiles);
}